// NEUROPULSBell_Arct_44530220925627
// MI455X (gfx1250) — compile-verified
//
#include <hip/hip_runtime.h>
#include <math.h>

typedef __attribute__((ext_vector_type(2))) float v2f;
typedef __attribute__((ext_vector_type(8))) float v8f;

#define NPAIRS 256
#define NSTEPS 256
#define NDIM   512

__global__ void NEUROPULS_zero_kernel(float* __restrict__ out, int n) {
  int i = blockIdx.x * blockDim.x + threadIdx.x;
  int stride = gridDim.x * blockDim.x;
  for (; i < n; i += stride) out[i] = 0.0f;
}

// One workgroup per group of 4 row-pairs. Parallel product-tree of 2x2 complex
// step-transfer matrices; tree combines done as 16x16x16 f32 GEMMs on the 4x4
// real embeddings (block-diagonal 16x16) via V_WMMA_F32_16X16X4_F32.
__global__ __launch_bounds__(256) void NEUROPULS_mesh_kernel(
    const float* __restrict__ in_phase,    // (512)
    const float* __restrict__ out_phase,   // (512)
    const float* __restrict__ full_phases, // (256,512)
    const float* __restrict__ losses,      // (512,256)
    const float* __restrict__ imbs,        // (512,256)
    float* __restrict__ out)               // (512,512) complex64 interleaved
{
  // node[s] = compact 2x2 complex for 4 pairs: [step][pair][i*2+j][re/im]
  __shared__ float nodeLds[NSTEPS * 4 * 4 * 2]; // 32 KB

  const int g = blockIdx.x;   // pair group: pairs 4g..4g+3
  const int t = threadIdx.x;  // 0..255
  const int s = t;            // leaf step index

  // ---------------- leaves: T_s = B1 * diag(e^{i phi}) * B0 ----------------
  const float L2_10_OVER_20 = 0.16609640474436813f; // log2(10)/20
  #pragma unroll
  for (int p = 0; p < 4; ++p) {
    int k = 4 * g + p;
    float l0 = losses[(2 * s + 0) * NPAIRS + k];
    float l1 = losses[(2 * s + 1) * NPAIRS + k];
    float m0 = imbs  [(2 * s + 0) * NPAIRS + k];
    float m1 = imbs  [(2 * s + 1) * NPAIRS + k];
    float a0 = exp2f(-l0 * L2_10_OVER_20);        // 10^(-loss/20)
    float a1 = exp2f(-l1 * L2_10_OVER_20);
    float t0 = a0 * sqrtf((1.0f + m0) * 0.5f);
    float k0 = a0 * sqrtf((1.0f - m0) * 0.5f);
    float t1 = a1 * sqrtf((1.0f + m1) * 0.5f);
    float k1 = a1 * sqrtf((1.0f - m1) * 0.5f);
    float ph0 = full_phases[s * NDIM + 2 * k + 0];
    float ph1 = full_phases[s * NDIM + 2 * k + 1];
    float c0, s0, c1, s1;
    sincosf(ph0, &s0, &c0);
    sincosf(ph1, &s1, &c1);
    float tt = t1 * t0, kk = k1 * k0, tk = t1 * k0, kt = k1 * t0;
    float* nd = &nodeLds[((s * 4 + p) * 4) * 2];
    nd[0] = tt * c0 - kk * c1;      nd[1] = tt * s0 - kk * s1;      // T00
    nd[2] = -(tk * s0 + kt * s1);   nd[3] = tk * c0 + kt * c1;      // T01 = i(tk e0 + kt e1)
    nd[4] = -(kt * s0 + tk * s1);   nd[5] = kt * c0 + tk * c1;      // T10 = i(kt e0 + tk e1)
    nd[6] = -kk * c0 + tt * c1;     nd[7] = -kk * s0 + tt * s1;     // T11
  }

  const int wave = t >> 5;
  const int lane = t & 31;
  const int nn   = lane & 15;   // plays role of m for A, n for B/D
  const int h    = lane >> 4;   // lane-half
  const int blk  = nn >> 2;     // this lane's pair-block (same for A and B)

  // ---------------- product tree: node[b] <- node[b+st] * node[b] ----------
  for (int st = 1; st < NSTEPS; st <<= 1) {
    __syncthreads();
    int ncomb = NSTEPS / (2 * st);
    for (int c = wave; c < ncomb; c += 8) {
      int b = c * 2 * st;
      // A fragment (high segment). A[m][k]: lane = m + 16*(k/2 within chunk),
      // vgpr = k%2. Nonzero only in chunk == blk; z = Zhigh[blk][(m&3)>>1][h].
      const float* zAp = &nodeLds[(((b + st) * 4 + blk) * 4 + (((nn & 3) >> 1) * 2 + h)) * 2];
      float zAr = zAp[0], zAi = zAp[1];
      float a0 = (nn & 1) ? zAi : zAr;    // k-parity 0
      float a1 = (nn & 1) ? zAr : -zAi;   // k-parity 1
      // B fragment (low segment). B[k][n]: lane = n + 16*(k/2), vgpr = k%2.
      // z = Zlow[blk][h][(n&3)>>1].
      const float* zBp = &nodeLds[((b * 4 + blk) * 4 + (h * 2 + ((nn & 3) >> 1))) * 2];
      float zBr = zBp[0], zBi = zBp[1];
      float b0 = (nn & 1) ? -zBi : zBr;
      float b1 = (nn & 1) ? zBr : zBi;

      v8f d = {0.f, 0.f, 0.f, 0.f, 0.f, 0.f, 0.f, 0.f};
      #pragma unroll
      for (int c4 = 0; c4 < 4; ++c4) {
        v2f av, bv;
        av.x = (c4 == blk) ? a0 : 0.0f;
        av.y = (c4 == blk) ? a1 : 0.0f;
        bv.x = (c4 == blk) ? b0 : 0.0f;
        bv.y = (c4 == blk) ? b1 : 0.0f;
        d = __builtin_amdgcn_wmma_f32_16x16x4_f32(
            /*neg_a=*/false, av, /*neg_b=*/false, bv,
            /*c_mod=*/(short)0, d, /*reuse_a=*/false, /*reuse_b=*/false);
      }
      // Extract compact complex: Re z_ij = D[4p+2i][4p+2j], Im = D[4p+2i+1][..].
      // For pair-block p the rows 4p..4p+3 live in lane-half h == p>>1, and
      // the D vgpr index is 4*(p&1) + 2*i (+1 for Im) -- a constant 2-way
      // select instead of a dynamic vector subscript.
      if (((nn & 1) == 0) && ((blk >> 1) == h)) {
        int p  = blk;
        int j  = (nn & 3) >> 1;
        bool hi = (p & 1) != 0;
        float re0 = hi ? d[4] : d[0];   // i=0, Re
        float im0 = hi ? d[5] : d[1];   // i=0, Im
        float re1 = hi ? d[6] : d[2];   // i=1, Re
        float im1 = hi ? d[7] : d[3];   // i=1, Im
        float* op0 = &nodeLds[((b * 4 + p) * 4 + (0 * 2 + j)) * 2];
        op0[0] = re0; op0[1] = im0;
        float* op1 = &nodeLds[((b * 4 + p) * 4 + (1 * 2 + j)) * 2];
        op1[0] = re1; op1[1] = im1;
      }
    }
  }
  __syncthreads();

  // ------------- apply in/out heaters, scatter 2x2 blocks to output --------
  if (t < 16) {
    int p  = t >> 2;
    int ij = t & 3;
    int i2 = ij >> 1, j2 = ij & 1;
    int k   = 4 * g + p;
    int row = 2 * k + i2, col = 2 * k + j2;
    float pr = nodeLds[((0 * 4 + p) * 4 + ij) * 2 + 0];
    float pi = nodeLds[((0 * 4 + p) * 4 + ij) * 2 + 1];
    float si, ci, so, co;
    sincosf(in_phase[col], &si, &ci);
    sincosf(out_phase[row], &so, &co);
    float xr = pr * ci - pi * si;   // P * e^{i in}
    float xi = pr * si + pi * ci;
    float fr = xr * co - xi * so;   // e^{i out} * (...)
    float fi = xr * so + xi * co;
    out[(row * NDIM + col) * 2 + 0] = fr;
    out[(row * NDIM + col) * 2 + 1] = fi;
  }
}

extern "C" void kernel_launch(void* const* d_in, const int* in_sizes, int n_in,
                              void* d_out, int out_size, void* d_ws, size_t ws_size,
                              hipStream_t stream) {
  (void)in_sizes; (void)n_in; (void)d_ws; (void)ws_size;
  const float* ht_in   = (const float*)d_in[0];
  const float* ht_out  = (const float*)d_in[1];
  const float* ht_full = (const float*)d_in[2];
  const float* losses  = (const float*)d_in[3];
  const float* imbs    = (const float*)d_in[4];
  float* out = (float*)d_out;

  // Output is a 512x512 complex64 matrix = 524288 floats; zero everything
  // first (it is block-diagonal, almost all entries are exactly 0).
  int nfloats = out_size < (NDIM * NDIM * 2) ? (NDIM * NDIM * 2) : out_size;
  NEUROPULS_zero_kernel<<<256, 256, 0, stream>>>(out, nfloats);

  // 64 groups of 4 row-pairs, 256 threads (8 waves) each.
  NEUROPULS_mesh_kernel<<<64, 256, 0, stream>>>(ht_in, ht_out, ht_full,
                                                losses, imbs, out);
}